// HieLayer_66417374265543
// MI455X (gfx1250) — compile-verified
//
#include <hip/hip_runtime.h>
#include <hip/hip_bf16.h>

// ---------------------------------------------------------------------------
// Problem constants (match the reference).
// ---------------------------------------------------------------------------
#define HIE_B 32
#define HIE_T 4096
#define HIE_H 256
#define HIE_S 256   // MAX_SENTS

typedef float v2f __attribute__((ext_vector_type(2)));
typedef float v8f __attribute__((ext_vector_type(8)));

static __device__ __forceinline__ int imin(int a, int b) { return a < b ? a : b; }

// ---------------------------------------------------------------------------
// Phase 1: one wave32 per doc. Ballot/popcount scan of boundary flags to get
// sentence end offsets:  starts[b][0]=0, starts[b][s+1]=pos_of_boundary_s + 1.
// Also emits doc_lens (int for phase 2, float into the d_out tail).
// ---------------------------------------------------------------------------
__global__ __launch_bounds__(32) void hie_scan_kernel(const int* __restrict__ bx,
                                                      int* __restrict__ starts,
                                                      int* __restrict__ dlen,
                                                      float* __restrict__ dlen_out) {
  const int b    = blockIdx.x;
  const int lane = threadIdx.x;                 // 0..31 (wave32)
  int* st = starts + b * (HIE_S + 1);

  // Zero-init this doc's offset table (single wave -> program order is safe).
  for (int i = lane; i <= HIE_S; i += 32) st[i] = 0;

  const int* row = bx + b * HIE_T;
  int run = 0;
  for (int base = 0; base < HIE_T; base += 32) {
    const int t    = base + lane;
    const int flag = (row[t] == 1) ? 1 : 0;
    const unsigned m = (unsigned)__ballot(flag != 0);      // wave32: low 32 bits
    const int prefix = __popc(m & ((1u << lane) - 1u));
    if (flag) {
      const int idx = run + prefix;                        // boundary ordinal
      if (idx < HIE_S) st[idx + 1] = t + 1;                // sentence idx ends at t
    }
    run += __popc(m);
  }
  if (lane == 0) {
    dlen[b]     = imin(run, HIE_S);
    dlen_out[b] = (float)run;                              // reference doc_lens
  }
}

// ---------------------------------------------------------------------------
// Phase 2: segmented mean-pool as a one-hot f32 GEMM on the matrix pipe.
//   pooled_sums(16 sents x 16 cols) += A(16x4 one-hot) * B(4 tokens x 16 cols)
// Grid: B * (S/16) blocks; 16 waves/block, wave w owns columns [16w, 16w+16).
// Each block streams only the token band belonging to its 16 sentences, so
// every input element is read exactly once -> bandwidth-floor traffic.
// Main loop is unrolled x2 with dual accumulators and incremental addressing;
// the possibly-clamped tail chunks are peeled out of the hot loop.
// ---------------------------------------------------------------------------
__global__ __launch_bounds__(512) void hie_pool_wmma_kernel(const float* __restrict__ X,
                                                            const int* __restrict__ starts,
                                                            const int* __restrict__ dlen,
                                                            float* __restrict__ out) {
  const int b    = blockIdx.x >> 4;
  const int s0   = (blockIdx.x & 15) << 4;      // first sentence row of this tile
  const int wave = threadIdx.x >> 5;
  const int lane = threadIdx.x & 31;
  const int ncol = (wave << 4) + (lane & 15);   // H column owned by this lane

  const int  L  = dlen[b];
  const int* st = starts + b * (HIE_S + 1);
  const float* Xb = X + (size_t)b * HIE_T * HIE_H;

  // A-operand row owned by this lane (ISA f32 16x4 layout: lanes 0-15 hold
  // K=0,1 in v[0],v[1]; lanes 16-31 hold K=2,3). M = lane & 15.
  const int rowA = s0 + (lane & 15);
  const int stA  = (rowA < L) ? st[rowA]     : -1;
  const int enA  = (rowA < L) ? st[rowA + 1] : -1;
  const int kx   = (lane < 16) ? 0 : 2;

  // Token band for this 16-sentence tile (block-uniform => EXEC stays full).
  const int tb = (s0 < L) ? st[s0] : 0;
  const int te = (s0 < L) ? st[imin(s0 + 16, L)] : 0;

  v8f c0 = {0.f, 0.f, 0.f, 0.f, 0.f, 0.f, 0.f, 0.f};
  v8f c1 = c0;

  // Incrementally-addressed pointer for this lane's K rows (t_x, t_x+1).
  const float* px = Xb + (size_t)(tb + kx) * HIE_H + ncol;
  int t0 = tb;

  // Hot loop: 8 tokens (2 chunks) per iteration, no clamps, immediate-offset
  // loads off a single advancing base pointer.
  for (; t0 + 8 <= te; t0 += 8, px += 8 * HIE_H) {
    v2f a0, a1, b0, b1;
    b0.x = px[0];
    b0.y = px[HIE_H];
    b1.x = px[4 * HIE_H];
    b1.y = px[5 * HIE_H];

    __builtin_prefetch(px + 32 * HIE_H, 0, 0);             // speculative, safe

    const int tx = t0 + kx;
    a0.x = (tx     >= stA && tx     < enA) ? 1.0f : 0.0f;
    a0.y = (tx + 1 >= stA && tx + 1 < enA) ? 1.0f : 0.0f;
    a1.x = (tx + 4 >= stA && tx + 4 < enA) ? 1.0f : 0.0f;
    a1.y = (tx + 5 >= stA && tx + 5 < enA) ? 1.0f : 0.0f;

    c0 = __builtin_amdgcn_wmma_f32_16x16x4_f32(false, a0, false, b0,
                                               (short)0, c0, false, false);
    c1 = __builtin_amdgcn_wmma_f32_16x16x4_f32(false, a1, false, b1,
                                               (short)0, c1, false, false);
  }

  // Peeled tail: up to 2 chunks; only here can token indices pass T-1.
  for (; t0 < te; t0 += 4) {
    const int t_x = t0 + kx;
    const int t_y = t_x + 1;
    v2f a, bb;
    a.x = (t_x >= stA && t_x < enA) ? 1.0f : 0.0f;
    a.y = (t_y >= stA && t_y < enA) ? 1.0f : 0.0f;
    const int cx = imin(t_x, HIE_T - 1);
    const int cy = imin(t_y, HIE_T - 1);
    bb.x = Xb[(size_t)cx * HIE_H + ncol];
    bb.y = Xb[(size_t)cy * HIE_H + ncol];
    c0 = __builtin_amdgcn_wmma_f32_16x16x4_f32(false, a, false, bb,
                                               (short)0, c0, false, false);
  }

  const v8f c = c0 + c1;

  // 1/count per sentence row of this tile, via LDS.
  __shared__ float invcnt[16];
  if (threadIdx.x < 16) {
    const int r = s0 + threadIdx.x;
    const float cnt = (r < L) ? (float)(st[r + 1] - st[r]) : 1.0f;
    invcnt[threadIdx.x] = 1.0f / fmaxf(cnt, 1.0f);
  }
  __syncthreads();

  // C/D layout: VGPR j -> M=j (lanes 0-15) / M=j+8 (lanes 16-31), N=lane&15.
  const int half = (lane >> 4) << 3;
#pragma unroll
  for (int j = 0; j < 8; ++j) {
    const int m   = j + half;
    const int row = s0 + m;
    const float v = (row < L) ? c[j] * invcnt[m] : 0.0f;   // zero-pad invalid
    out[((size_t)b * HIE_S + row) * HIE_H + ncol] = v;
  }
}

// ---------------------------------------------------------------------------
// Launch: phase 1 (scan) then phase 2 (WMMA pool), same stream => ordered.
// Workspace: starts[B][S+1] ints + dlen[B] ints  (~33 KB).
// d_out: pooled (B*S*H floats) followed by doc_lens (B floats).
// ---------------------------------------------------------------------------
extern "C" void kernel_launch(void* const* d_in, const int* in_sizes, int n_in,
                              void* d_out, int out_size, void* d_ws, size_t ws_size,
                              hipStream_t stream) {
  (void)in_sizes; (void)n_in; (void)out_size; (void)ws_size;

  const float* X  = (const float*)d_in[0];    // hie_ins  (B,T,H) f32
  const int*   bx = (const int*)d_in[1];      // batch_x  (B,T)   i32
  float* out = (float*)d_out;

  int* starts = (int*)d_ws;
  int* dlen   = starts + HIE_B * (HIE_S + 1);
  float* dlen_out = out + (size_t)HIE_B * HIE_S * HIE_H;

  hie_scan_kernel<<<HIE_B, 32, 0, stream>>>(bx, starts, dlen, dlen_out);
  hie_pool_wmma_kernel<<<HIE_B * (HIE_S / 16), 512, 0, stream>>>(X, starts, dlen, out);
}